// SparseGlobalAvgPool2d_90984587198527
// MI455X (gfx1250) — compile-verified
//
#include <hip/hip_runtime.h>

// ---------------------------------------------------------------------------
// SparseGlobalAvgPool2d: column-sum of [N=2e6, C=64] f32, scaled by 1/(h*w).
// HBM-bound: 512 MB stream -> ~22us floor at 23.3 TB/s.
// Hot loop: coalesced global_load_b128 + per-lane float4 accumulation.
// Epilogue: per-block 16x64 partial matrix reduced on the matrix pipe with
//           V_WMMA_F32_16X16X4_F32 (A = ones), then global_atomic_add_f32.
// ---------------------------------------------------------------------------

typedef float v8f __attribute__((ext_vector_type(8)));
typedef float v2f __attribute__((ext_vector_type(2)));

#define NCHAN   64
#define BLOCK   256
#define GRID    1024
#define UNROLL  4

__global__ void sgap_zero_ws(float* __restrict__ ws) {
    ws[threadIdx.x] = 0.0f;
}

__global__ __launch_bounds__(BLOCK) void sgap_main(const float* __restrict__ feat,
                                                   float* __restrict__ ws,
                                                   int total4) {
    const float4* __restrict__ f4 = (const float4*)feat;

    // Grid-stride over float4s. All strides are multiples of 16 float4s
    // (one full 64-channel row), so each thread's 4 accumulated lanes stay
    // bound to the same 4 channels: channels 4*(tid%16) .. +3.
    int base   = blockIdx.x * (BLOCK * UNROLL) + threadIdx.x;
    int stride = gridDim.x * (BLOCK * UNROLL);

    float4 acc = make_float4(0.f, 0.f, 0.f, 0.f);
    for (int i = base; i < total4; i += stride) {
#pragma unroll
        for (int u = 0; u < UNROLL; ++u) {
            int j = i + u * BLOCK;           // +BLOCK keeps (j % 16) invariant
            if (j < total4) {
                float4 v = f4[j];
                acc.x += v.x; acc.y += v.y; acc.z += v.z; acc.w += v.w;
            }
        }
    }

    // Deposit per-thread partials: LDS matrix [16 row-groups][64 channels].
    __shared__ float lds[16 * NCHAN];
    int rg = threadIdx.x >> 4;          // row-group 0..15
    int cg = (threadIdx.x & 15) << 2;   // channel base 0..60
    *(float4*)&lds[rg * NCHAN + cg] = acc;
    __syncthreads();

    // Reduce the 16 row-groups per channel on the matrix pipe:
    // D = ones(16x4) x B(4x16) + C, chained over 4 K-tiles.
    // Waves 0..3 each own a 16-channel slice; EXEC is all-ones per wave.
    if (threadIdx.x < 128) {
        int wave = threadIdx.x >> 5;
        int lane = threadIdx.x & 31;
        int half = lane >> 4;                      // K half select (ISA B layout)
        int ch   = wave * 16 + (lane & 15);        // this lane's channel (N index)

        v2f a = {1.0f, 1.0f};                      // all-ones A matrix (16x4)
        v8f d = {0.f, 0.f, 0.f, 0.f, 0.f, 0.f, 0.f, 0.f};
#pragma unroll
        for (int kk = 0; kk < 4; ++kk) {
            // B(4x16) f32 layout: lanes 0-15 carry K=0,1 in VGPR0/1 (N=lane),
            // lanes 16-31 carry K=2,3. Map K -> row-group kk*4 + 2*half + {0,1}.
            v2f b;
            b.x = lds[(kk * 4 + 2 * half + 0) * NCHAN + ch];
            b.y = lds[(kk * 4 + 2 * half + 1) * NCHAN + ch];
            d = __builtin_amdgcn_wmma_f32_16x16x4_f32(
                    /*neg_a=*/false, a, /*neg_b=*/false, b,
                    /*c_mod=*/(short)0, d, /*reuse_a=*/false, /*reuse_b=*/false);
        }
        // Every row of D equals the column sum; take M=0 (VGPR0, lanes 0-15).
        if (half == 0) {
            unsafeAtomicAdd(&ws[ch], d[0]);        // global_atomic_add_f32
        }
    }
}

__global__ void sgap_final(const float* __restrict__ ws,
                           const int* __restrict__ hp,
                           const int* __restrict__ wp,
                           float* __restrict__ out) {
    int c = threadIdx.x;                            // 64 threads
    float inv_size = 1.0f / ((float)hp[0] * (float)wp[0]);
    out[c] = ws[c] * inv_size;
}

extern "C" void kernel_launch(void* const* d_in, const int* in_sizes, int n_in,
                              void* d_out, int out_size, void* d_ws, size_t ws_size,
                              hipStream_t stream) {
    const float* feat = (const float*)d_in[0];   // [N, 64] f32
    const int*   hp   = (const int*)d_in[1];     // scalar h
    const int*   wp   = (const int*)d_in[2];     // scalar w
    float*       out  = (float*)d_out;           // [64] f32
    float*       ws   = (float*)d_ws;            // 64-float accumulator

    int total4 = in_sizes[0] >> 2;               // N*C/4 float4s (fits in int)

    sgap_zero_ws<<<1, NCHAN, 0, stream>>>(ws);
    sgap_main<<<GRID, BLOCK, 0, stream>>>(feat, ws, total4);
    sgap_final<<<1, NCHAN, 0, stream>>>(ws, hp, wp, out);
}